// KarateGraphLin1_68599217652364
// MI455X (gfx1250) — compile-verified
//
#include <hip/hip_runtime.h>
#include <hip/hip_bf16.h>
#include <stdint.h>

#define IN_F   256
#define HID    128
#define OUTC   16

typedef __attribute__((ext_vector_type(2))) float v2f;
typedef __attribute__((ext_vector_type(8))) float v8f;

// ---------------- degree / norm ----------------
__global__ void k_deg_init(float* deg, int N) {
    int i = blockIdx.x * blockDim.x + threadIdx.x;
    if (i < N) deg[i] = 1.0f;                       // self-loop contributes 1
}

__global__ void k_deg_count(const long long* dst, float* deg, int E) {
    int e = blockIdx.x * blockDim.x + threadIdx.x;
    if (e < E) atomicAdd(&deg[(int)dst[e]], 1.0f);
}

__global__ void k_deg_rsqrt(float* deg, int N) {
    int i = blockIdx.x * blockDim.x + threadIdx.x;
    if (i < N) {
        float d = deg[i];
        deg[i] = d > 0.0f ? __frsqrt_rn(d) : 0.0f;  // now holds dinv
    }
}

// ---------------- GEMM1: h = x @ W_conv  (N x 256) @ (256 x 128) ----------------
// Block = 256 threads = 8 waves, computing a 32-row x 128-col output tile.
// Each wave owns cols [16w,16w+16) and TWO 16x16 M-tiles (rows +0 and +16):
// the B fragment is reused across both -> 2 WMMAs per B load pair.
#define G1_ROWS 32
__global__ void k_gemm1_wmma(const float* __restrict__ x,
                             const float* __restrict__ Wc,
                             float* __restrict__ h, int N) {
    __shared__ float sA[G1_ROWS][IN_F + 1];         // +1 pad: conflict-free column reads
    const int row0 = blockIdx.x * G1_ROWS;
    const bool full = (row0 + G1_ROWS) <= N;

    // cooperative load of 32x256 fp32 A-tile (2048 float4, 256 threads x 8)
    if (full) {
        #pragma unroll
        for (int i = 0; i < 8; ++i) {
            int q  = threadIdx.x + i * 256;         // float4 index in [0,2048)
            int r  = q >> 6;                         // 64 float4 per row
            int c4 = q & 63;
            float4 v = ((const float4*)(x + (size_t)(row0 + r) * IN_F))[c4];
            sA[r][c4 * 4 + 0] = v.x;
            sA[r][c4 * 4 + 1] = v.y;
            sA[r][c4 * 4 + 2] = v.z;
            sA[r][c4 * 4 + 3] = v.w;
        }
    } else {
        #pragma unroll
        for (int i = 0; i < 8; ++i) {
            int q  = threadIdx.x + i * 256;
            int r  = q >> 6;
            int c4 = q & 63;
            float4 v = make_float4(0.f, 0.f, 0.f, 0.f);
            if (row0 + r < N) v = ((const float4*)(x + (size_t)(row0 + r) * IN_F))[c4];
            sA[r][c4 * 4 + 0] = v.x;
            sA[r][c4 * 4 + 1] = v.y;
            sA[r][c4 * 4 + 2] = v.z;
            sA[r][c4 * 4 + 3] = v.w;
        }
    }
    __syncthreads();

    const int wave = threadIdx.x >> 5;
    const int lane = threadIdx.x & 31;
    const int half = lane >> 4;                     // 0: K={0,1}  1: K={2,3}
    const int l    = lane & 15;
    const int n0   = wave * 16;                     // column tile
    const float* wp = Wc + n0 + l;

    v8f acc0 = {};
    v8f acc1 = {};
    #pragma unroll 4
    for (int k0 = 0; k0 < IN_F; k0 += 4) {
        int ka = k0 + 2 * half;
        v2f a0, a1, b;
        a0[0] = sA[l][ka + 0];
        a0[1] = sA[l][ka + 1];
        a1[0] = sA[16 + l][ka + 0];
        a1[1] = sA[16 + l][ka + 1];
        b[0]  = wp[(size_t)(ka + 0) * HID];
        b[1]  = wp[(size_t)(ka + 1) * HID];
        acc0 = __builtin_amdgcn_wmma_f32_16x16x4_f32(false, a0, false, b,
                                                     (short)0, acc0, false, false);
        acc1 = __builtin_amdgcn_wmma_f32_16x16x4_f32(false, a1, false, b,
                                                     (short)0, acc1, false, false);
    }

    // C layout: VGPR r -> M = r + 8*(lane>=16), N = lane&15
    float* hp = h + (size_t)(row0 + 8 * half) * HID + n0 + l;
    if (full) {
        #pragma unroll
        for (int r = 0; r < 8; ++r) {
            hp[(size_t)r * HID]        = acc0[r];
            hp[(size_t)(r + 16) * HID] = acc1[r];
        }
    } else {
        #pragma unroll
        for (int r = 0; r < 8; ++r) {
            int ra = row0 + r + 8 * half;
            int rb = ra + 16;
            if (ra < N) h[(size_t)ra * HID + n0 + l] = acc0[r];
            if (rb < N) h[(size_t)rb * HID + n0 + l] = acc1[r];
        }
    }
}

// ---------------- self-loop init: agg[i] = h[i] * dinv[i]^2 ----------------
__global__ void k_selfloop(const float* __restrict__ h, const float* __restrict__ dinv,
                           float* __restrict__ agg, int N) {
    size_t t = (size_t)blockIdx.x * blockDim.x + threadIdx.x;
    if (t < (size_t)N * HID) {
        int i = (int)(t >> 7);                      // /128
        float di = dinv[i];
        agg[t] = h[t] * di * di;
    }
}

// ---------------- edge scatter: agg[dst] += h[src] * dinv[src]*dinv[dst] ----------------
// One wave per edge; lane handles 4 floats: float4 gather + 4 L2-resident atomics.
__global__ void k_scatter(const long long* __restrict__ src,
                          const long long* __restrict__ dst,
                          const float* __restrict__ h,
                          const float* __restrict__ dinv,
                          float* __restrict__ agg, int E) {
    int wid  = (blockIdx.x * blockDim.x + threadIdx.x) >> 5;
    int lane = threadIdx.x & 31;
    if (wid >= E) return;
    int s = (int)src[wid];
    int d = (int)dst[wid];
    float nrm = dinv[s] * dinv[d];
    float4 v  = ((const float4*)(h + (size_t)s * HID))[lane];
    float* ap = agg + (size_t)d * HID + lane * 4;
    atomicAdd(ap + 0, v.x * nrm);
    atomicAdd(ap + 1, v.y * nrm);
    atomicAdd(ap + 2, v.z * nrm);
    atomicAdd(ap + 3, v.w * nrm);
}

// ---------------- GEMM2 + bias + relu + log_softmax ----------------
// Block = 128 threads = 4 waves; each wave handles 16 nodes (one 16x16 WMMA C tile,
// K=128 -> 32x V_WMMA_F32_16X16X4_F32). Softmax over 16 cols = 16-lane shuffle tree.
#define W7 4
__global__ void k_gemm2_softmax(const float* __restrict__ agg,
                                const float* __restrict__ bconv,
                                const float* __restrict__ Wl,
                                const float* __restrict__ bl,
                                float* __restrict__ out, int N) {
    __shared__ float sT[W7][16][HID + 4];           // pad 4: bank stride 4, conflict-free
    const int wave = threadIdx.x >> 5;
    const int lane = threadIdx.x & 31;
    const int row0 = blockIdx.x * (W7 * 16) + wave * 16;
    const bool full = (row0 + 16) <= N;

    // per-wave cooperative load of 16x128 tile with fused +b_conv and relu
    float4 bb = ((const float4*)bconv)[lane];
    #pragma unroll
    for (int r = 0; r < 16; ++r) {
        float4 v = make_float4(0.f, 0.f, 0.f, 0.f);
        if (full || (row0 + r < N))
            v = ((const float4*)(agg + (size_t)(row0 + r) * HID))[lane];
        sT[wave][r][lane * 4 + 0] = fmaxf(v.x + bb.x, 0.f);
        sT[wave][r][lane * 4 + 1] = fmaxf(v.y + bb.y, 0.f);
        sT[wave][r][lane * 4 + 2] = fmaxf(v.z + bb.z, 0.f);
        sT[wave][r][lane * 4 + 3] = fmaxf(v.w + bb.w, 0.f);
    }
    __syncthreads();

    const int half = lane >> 4;
    const int l    = lane & 15;

    v8f acc = {};
    #pragma unroll 8
    for (int k0 = 0; k0 < HID; k0 += 4) {
        int ka = k0 + 2 * half;
        v2f a, b;
        a[0] = sT[wave][l][ka + 0];
        a[1] = sT[wave][l][ka + 1];
        b[0] = Wl[(ka + 0) * OUTC + l];
        b[1] = Wl[(ka + 1) * OUTC + l];
        acc = __builtin_amdgcn_wmma_f32_16x16x4_f32(false, a, false, b,
                                                    (short)0, acc, false, false);
    }

    float blin = bl[l];
    float* op = out + (size_t)(row0 + 8 * half) * OUTC + l;
    #pragma unroll
    for (int r = 0; r < 8; ++r) {
        float val = acc[r] + blin;
        // row M = r + 8*half lives across lanes [16*half, 16*half+16): reduce width 16
        float mx = val;
        #pragma unroll
        for (int m = 1; m < 16; m <<= 1) mx = fmaxf(mx, __shfl_xor(mx, m, 16));
        float ex = __expf(val - mx);
        float sum = ex;
        #pragma unroll
        for (int m = 1; m < 16; m <<= 1) sum += __shfl_xor(sum, m, 16);
        float res = val - mx - __logf(sum);
        if (full) {
            op[r * OUTC] = res;
        } else {
            int row = row0 + r + 8 * half;
            if (row < N) out[(size_t)row * OUTC + l] = res;
        }
    }
}

extern "C" void kernel_launch(void* const* d_in, const int* in_sizes, int n_in,
                              void* d_out, int out_size, void* d_ws, size_t ws_size,
                              hipStream_t stream) {
    const float*     x    = (const float*)d_in[0];
    const long long* eidx = (const long long*)d_in[1];   // int64 (2, E) row-major
    const float*     Wc   = (const float*)d_in[2];       // (256, 128)
    const float*     bc   = (const float*)d_in[3];       // (128,)
    const float*     Wl   = (const float*)d_in[4];       // (128, 16)
    const float*     bl   = (const float*)d_in[5];       // (16,)
    float*           out  = (float*)d_out;

    const int N = in_sizes[0] / IN_F;
    const int E = in_sizes[1] / 2;
    const long long* src = eidx;
    const long long* dst = eidx + E;

    // workspace: dinv [N] | h [N*128] | agg [N*128]
    float* dinv = (float*)d_ws;
    size_t off  = (((size_t)N * sizeof(float)) + 255) & ~(size_t)255;
    float* h    = (float*)((char*)d_ws + off);
    float* agg  = h + (size_t)N * HID;

    // 1) degrees -> dinv
    k_deg_init <<<(N + 255) / 256, 256, 0, stream>>>(dinv, N);
    k_deg_count<<<(E + 255) / 256, 256, 0, stream>>>(dst, dinv, E);
    k_deg_rsqrt<<<(N + 255) / 256, 256, 0, stream>>>(dinv, N);

    // 2) h = x @ W_conv   (WMMA fp32, 32-row blocks, B-fragment reuse)
    k_gemm1_wmma<<<(N + G1_ROWS - 1) / G1_ROWS, 256, 0, stream>>>(x, Wc, h, N);

    // 3) agg = self-loop term, then scatter-add edge messages (L2-resident atomics)
    {
        size_t total = (size_t)N * HID;
        k_selfloop<<<(unsigned)((total + 255) / 256), 256, 0, stream>>>(h, dinv, agg, N);
    }
    k_scatter<<<(E * 32 + 255) / 256, 256, 0, stream>>>(src, dst, h, dinv, agg, E);

    // 4) relu(agg + b_conv) @ W_lin + b_lin -> log_softmax  (WMMA fp32 + shuffle softmax)
    k_gemm2_softmax<<<(N + W7 * 16 - 1) / (W7 * 16), W7 * 32, 0, stream>>>(
        agg, bc, Wl, bl, out, N);
}